// DecoderModel_60696477827149
// MI455X (gfx1250) — compile-verified
//
#include <hip/hip_runtime.h>
#include <math.h>

typedef __attribute__((ext_vector_type(2))) float v2f;
typedef __attribute__((ext_vector_type(8))) float v8f;

#define NN    2048      // nodes
#define BB    32        // batch
#define UU    64        // units
#define HXSZ  (NN*UU)   // 131072 per batch row
#define ROWS  (BB*NN)   // 65536

#define BM 128
#define BN2 128         // gemm_cheb N tile
#define BN 64           // gemm_mix N tile
#define KB 32
#define PITCH 36        // LDS row pitch (floats): even, 16B-storable, conflict-free frags

#define CP0 2176        // cell0 x pitch (65*32=2080 padded to /128)
#define CP1 4096        // cell1 x pitch (128*32)

// ---------------- support construction ----------------
__global__ void rowsum_kernel(const float* __restrict__ adj, float* __restrict__ rs) {
  __shared__ float red[256];
  int r = blockIdx.x;
  float s = 0.f;
  for (int j = threadIdx.x; j < NN; j += 256) s += adj[r * NN + j];
  red[threadIdx.x] = s;
  __syncthreads();
  for (int off = 128; off > 0; off >>= 1) {
    if (threadIdx.x < off) red[threadIdx.x] += red[threadIdx.x + off];
    __syncthreads();
  }
  if (threadIdx.x == 0) rs[r] = red[0];
}

__global__ void colsum_kernel(const float* __restrict__ adj, float* __restrict__ cs) {
  int j = blockIdx.x * 256 + threadIdx.x;
  float s = 0.f;
  for (int i = 0; i < NN; ++i) s += adj[i * NN + j];
  cs[j] = s;
}

// S0[i][j] = adj[j][i]/max(rowsum[j],eps);  S1[i][j] = adj[i][j]/max(colsum[j],eps)
__global__ void supports_kernel(const float* __restrict__ adj,
                                const float* __restrict__ rs,
                                const float* __restrict__ cs,
                                float* __restrict__ S0, float* __restrict__ S1) {
  int j = blockIdx.x * 256 + threadIdx.x;
  int i = blockIdx.y;
  float inv_r = 1.f / fmaxf(rs[j], 1e-8f);
  float inv_c = 1.f / fmaxf(cs[j], 1e-8f);
  S0[i * NN + j] = adj[j * NN + i] * inv_r;
  S1[i * NN + j] = adj[i * NN + j] * inv_c;
}

// ---------------- x0 packing: x0[n, f*B+b] = cat[b,n,f] ----------------
__global__ void pack_c0(const float* __restrict__ inputs, const float* __restrict__ hx,
                        const float* __restrict__ gate, int use_r,
                        float* __restrict__ x0) {
  int c = blockIdx.x * 256 + threadIdx.x;
  int n = blockIdx.y;
  if (c >= CP0) return;
  float v = 0.f;
  if (c < 65 * BB) {
    int f = c >> 5, b = c & 31;
    if (f == 0) v = inputs[b * NN + n];
    else {
      int u = f - 1;
      v = hx[b * HXSZ + n * UU + u];
      if (use_r) v *= gate[(size_t)(b * NN + n) * 128 + u];
    }
  }
  x0[(size_t)n * CP0 + c] = v;
}

__global__ void pack_c1(const float* __restrict__ inp, const float* __restrict__ hx,
                        const float* __restrict__ gate, int use_r,
                        float* __restrict__ x0) {
  int c = blockIdx.x * 256 + threadIdx.x;   // 4096 cols exactly
  int n = blockIdx.y;
  int f = c >> 5, b = c & 31;
  float v;
  if (f < UU) {
    v = inp[b * HXSZ + n * UU + f];
  } else {
    int u = f - UU;
    v = hx[b * HXSZ + n * UU + u];
    if (use_r) v *= gate[(size_t)(b * NN + n) * 128 + u];
  }
  x0[(size_t)n * CP1 + c] = v;
}

// ---------------- fp32 WMMA GEMM: Y = alpha*(S @ X) + beta*Z ----------------
// S: 2048x2048, X/Z/Y: 2048 x Cp (row-major, Cp multiple of 128)
// 8 waves: wave w owns rows [w*16, w*16+16) x 128 cols (8 accumulators).
// S tile staged via CDNA5 async LDS copy (ASYNCcnt); X tile transposed in LDS
// so both A and B fragments are contiguous ds_load_b64.
__global__ __launch_bounds__(256)
void gemm_cheb(const float* __restrict__ S, const float* __restrict__ X,
               const float* __restrict__ Z, float* __restrict__ Y,
               int Cp, float alpha, float beta) {
  __shared__ float sS[BM][PITCH];     // [row][k]
  __shared__ float sXT[BN2][PITCH];   // [col][k]
  const int tid  = threadIdx.x;
  const int wave = tid >> 5;
  const int lane = tid & 31;
  const int half = lane >> 4;
  const int lr   = lane & 15;
  const int m_blk = blockIdx.y * BM;
  const int n_blk = blockIdx.x * BN2;
  const int m0 = wave * 16;

  // hoisted per-thread fill coordinates (no per-iter 64-bit muls)
  const float* sgp[4]; unsigned slp[4];
  const float* xgp[4]; int xr[4], xc[4];
#pragma unroll
  for (int q = 0; q < 4; ++q) {
    int cid = q * 256 + tid;
    int r = cid >> 3, c4 = cid & 7;          // S tile: 128 rows x 8 16B-chunks
    sgp[q] = S + (size_t)(m_blk + r) * NN + c4 * 4;
    slp[q] = (unsigned)(size_t)&sS[r][c4 * 4];
    int r2 = cid >> 5, c42 = cid & 31;       // X tile: 32 rows x 32 16B-chunks
    xgp[q] = X + (size_t)r2 * Cp + n_blk + c42 * 4;
    xr[q] = r2; xc[q] = c42 * 4;
  }

  v8f acc[8];
#pragma unroll
  for (int t = 0; t < 8; ++t) acc[t] = (v8f){};

  for (int kb = 0; kb < NN; kb += KB) {
    // S tile: global -> LDS, no VGPR round trip (async, 16B per lane-op)
#pragma unroll
    for (int q = 0; q < 4; ++q) {
      asm volatile("global_load_async_to_lds_b128 %0, %1, off"
                   :: "v"(slp[q]), "v"(sgp[q] + kb) : "memory");
    }
    // X tile: b128 loads, transposed scalar stores
#pragma unroll
    for (int q = 0; q < 4; ++q) {
      float4 v = *(const float4*)xgp[q];
      int c = xc[q], r = xr[q];
      sXT[c + 0][r] = v.x; sXT[c + 1][r] = v.y;
      sXT[c + 2][r] = v.z; sXT[c + 3][r] = v.w;
      xgp[q] += (size_t)KB * Cp;
    }
    asm volatile("s_wait_asynccnt 0x0" ::: "memory");
    __syncthreads();
#pragma unroll
    for (int kk = 0; kk < KB; kk += 4) {
      // A frag: lanes 0-15 K=kk,kk+1 ; lanes 16-31 K=kk+2,kk+3
      v2f a = *(const v2f*)&sS[m0 + lr][kk + 2 * half];
#pragma unroll
      for (int t = 0; t < 8; ++t) {
        v2f b = *(const v2f*)&sXT[t * 16 + lr][kk + 2 * half];
        acc[t] = __builtin_amdgcn_wmma_f32_16x16x4_f32(
            false, a, false, b, (short)0, acc[t], false, false);
      }
    }
    __syncthreads();
  }
#pragma unroll
  for (int t = 0; t < 8; ++t) {
#pragma unroll
    for (int r = 0; r < 8; ++r) {           // D: vgpr r -> rows m0+r / m0+r+8
      int row = m_blk + m0 + r + 8 * half;
      int col = n_blk + t * 16 + lr;
      float v = alpha * acc[t][r];
      if (beta != 0.f) v += beta * Z[(size_t)row * Cp + col];
      Y[(size_t)row * Cp + col] = v;
    }
  }
}

// ---------------- WMMA GEMM: Y2 = act( Xbig @ W + bias ) ----------------
// Xbig[row=b*N+n, k=f*5+m] = xs[m][n*Cp + f*B + b]  (gathered into LDS on the fly)
__global__ __launch_bounds__(256)
void gemm_mix(const float* __restrict__ xs0, const float* __restrict__ xs1,
              const float* __restrict__ xs2, const float* __restrict__ xs3,
              const float* __restrict__ xs4,
              const float* __restrict__ w, const float* __restrict__ bias,
              float* __restrict__ Y, int F, int Cp, int Nout, int act) {
  __shared__ float sA[BM][PITCH];     // [row][k]
  __shared__ float sWT[BN][PITCH];    // [col][k]
  const int tid  = threadIdx.x;
  const int wave = tid >> 5;
  const int lane = tid & 31;
  const int half = lane >> 4;
  const int lr   = lane & 15;
  const int row_blk = blockIdx.y * BM;
  const int col_blk = blockIdx.x * BN;
  const int m0 = wave * 16;
  const int K = 5 * F;
  const int Kpad = (K + KB - 1) / KB * KB;

  v8f acc[4];
#pragma unroll
  for (int t = 0; t < 4; ++t) acc[t] = (v8f){};

  for (int kb = 0; kb < Kpad; kb += KB) {
#pragma unroll
    for (int e = 0; e < 16; ++e) {          // gather 128x32 Xbig tile
      int idx = e * 256 + tid;
      int rr = idx >> 5, kc = idx & 31;
      int row = row_blk + rr;
      int b = row >> 11, n = row & (NN - 1);
      int k = kb + kc;
      float v = 0.f;
      if (k < K) {
        int f = k / 5, m = k % 5;
        const float* src = (m == 0) ? xs0 : (m == 1) ? xs1 : (m == 2) ? xs2
                         : (m == 3) ? xs3 : xs4;
        v = src[(size_t)n * Cp + f * BB + b];
      }
      sA[rr][kc] = v;
    }
#pragma unroll
    for (int e = 0; e < 8; ++e) {           // 32x64 W tile, transposed store
      int idx = e * 256 + tid;
      int r = idx >> 6, c = idx & 63;
      int k = kb + r;
      sWT[c][r] = (k < K) ? w[(size_t)k * Nout + col_blk + c] : 0.f;
    }
    __syncthreads();
#pragma unroll
    for (int kk = 0; kk < KB; kk += 4) {
      v2f a = *(const v2f*)&sA[m0 + lr][kk + 2 * half];
#pragma unroll
      for (int t = 0; t < 4; ++t) {
        v2f b = *(const v2f*)&sWT[t * 16 + lr][kk + 2 * half];
        acc[t] = __builtin_amdgcn_wmma_f32_16x16x4_f32(
            false, a, false, b, (short)0, acc[t], false, false);
      }
    }
    __syncthreads();
  }
#pragma unroll
  for (int t = 0; t < 4; ++t) {
#pragma unroll
    for (int r = 0; r < 8; ++r) {
      int row = row_blk + m0 + r + 8 * half;
      int col = col_blk + t * 16 + lr;
      float v = acc[t][r] + bias[col];
      if (act == 1)      v = 1.f / (1.f + __expf(-v));
      else if (act == 2) v = tanhf(v);
      Y[(size_t)row * Nout + col] = v;
    }
  }
}

// ---------------- GRU combine: h = u*hx + (1-u)*c ----------------
__global__ void combine_h(const float* __restrict__ gate, const float* __restrict__ cand,
                          const float* __restrict__ hx, float* __restrict__ h_out) {
  int i = blockIdx.x * 256 + threadIdx.x;   // over B*N*UU
  int u = i & 63;
  int row = i >> 6;                         // b*N+n
  int b = row >> 11, n = row & (NN - 1);
  float ug = gate[(size_t)row * 128 + 64 + u];
  float hv = hx[b * HXSZ + n * UU + u];
  float c  = cand[(size_t)row * UU + u];
  h_out[b * HXSZ + n * UU + u] = ug * hv + (1.f - ug) * c;
}

// ---------------- projection: out = h1 @ w_proj + b_proj ----------------
__global__ void proj_kernel(const float* __restrict__ h1, const float* __restrict__ wp,
                            const float* __restrict__ bp, float* __restrict__ out) {
  int i = blockIdx.x * 256 + threadIdx.x;   // b*N+n
  int b = i >> 11, n = i & (NN - 1);
  const float* hrow = h1 + b * HXSZ + n * UU;
  float s = bp[0];
#pragma unroll
  for (int u = 0; u < UU; ++u) s += hrow[u] * wp[u];
  out[i] = s;
}

extern "C" void kernel_launch(void* const* d_in, const int* in_sizes, int n_in,
                              void* d_out, int out_size, void* d_ws, size_t ws_size,
                              hipStream_t stream) {
  const float* inputs = (const float*)d_in[0];
  const float* hidden = (const float*)d_in[1];
  const float* adj    = (const float*)d_in[2];
  const float* wg0 = (const float*)d_in[3];
  const float* bg0 = (const float*)d_in[4];
  const float* wc0 = (const float*)d_in[5];
  const float* bc0 = (const float*)d_in[6];
  const float* wg1 = (const float*)d_in[7];
  const float* bg1 = (const float*)d_in[8];
  const float* wc1 = (const float*)d_in[9];
  const float* bc1 = (const float*)d_in[10];
  const float* wp  = (const float*)d_in[11];
  const float* bp  = (const float*)d_in[12];

  float* out = (float*)d_out;               // [65536] out | [4M] h0 | [4M] h1
  float* h0  = out + ROWS;
  float* h1  = h0 + (size_t)BB * HXSZ;
  const float* hx0 = hidden;
  const float* hx1 = hidden + (size_t)BB * HXSZ;

  float* ws = (float*)d_ws;
  size_t off = 0;
  float* S0 = ws + off; off += (size_t)NN * NN;
  float* S1 = ws + off; off += (size_t)NN * NN;
  float* rs = ws + off; off += NN;
  float* cs = ws + off; off += NN;
  float* xs[5];
  for (int i = 0; i < 5; ++i) { xs[i] = ws + off; off += (size_t)NN * CP1; }
  float* gate = ws + off; off += (size_t)ROWS * 128;
  float* cand = ws + off; off += (size_t)ROWS * UU;

  // supports
  rowsum_kernel<<<NN, 256, 0, stream>>>(adj, rs);
  colsum_kernel<<<NN / 256, 256, 0, stream>>>(adj, cs);
  supports_kernel<<<dim3(NN / 256, NN), 256, 0, stream>>>(adj, rs, cs, S0, S1);

  auto cheb = [&](int Cp) {
    dim3 g(Cp / BN2, NN / BM);
    gemm_cheb<<<g, 256, 0, stream>>>(S0, xs[0], xs[0], xs[1], Cp, 1.f,  0.f);
    gemm_cheb<<<g, 256, 0, stream>>>(S0, xs[1], xs[0], xs[2], Cp, 2.f, -1.f);
    gemm_cheb<<<g, 256, 0, stream>>>(S1, xs[0], xs[0], xs[3], Cp, 1.f,  0.f);
    gemm_cheb<<<g, 256, 0, stream>>>(S1, xs[3], xs[0], xs[4], Cp, 2.f, -1.f);
  };

  // ---- cell 0 ----  F=65, Cp=2176
  pack_c0<<<dim3(9, NN), 256, 0, stream>>>(inputs, hx0, gate, 0, xs[0]);
  cheb(CP0);
  gemm_mix<<<dim3(2, ROWS / BM), 256, 0, stream>>>(xs[0], xs[1], xs[2], xs[3], xs[4],
      wg0, bg0, gate, 65, CP0, 128, 1);
  pack_c0<<<dim3(9, NN), 256, 0, stream>>>(inputs, hx0, gate, 1, xs[0]);
  cheb(CP0);
  gemm_mix<<<dim3(1, ROWS / BM), 256, 0, stream>>>(xs[0], xs[1], xs[2], xs[3], xs[4],
      wc0, bc0, cand, 65, CP0, 64, 2);
  combine_h<<<(BB * HXSZ) / 256, 256, 0, stream>>>(gate, cand, hx0, h0);

  // ---- cell 1 ----  F=128, Cp=4096
  pack_c1<<<dim3(16, NN), 256, 0, stream>>>(h0, hx1, gate, 0, xs[0]);
  cheb(CP1);
  gemm_mix<<<dim3(2, ROWS / BM), 256, 0, stream>>>(xs[0], xs[1], xs[2], xs[3], xs[4],
      wg1, bg1, gate, 128, CP1, 128, 1);
  pack_c1<<<dim3(16, NN), 256, 0, stream>>>(h0, hx1, gate, 1, xs[0]);
  cheb(CP1);
  gemm_mix<<<dim3(1, ROWS / BM), 256, 0, stream>>>(xs[0], xs[1], xs[2], xs[3], xs[4],
      wc1, bc1, cand, 128, CP1, 64, 2);
  combine_h<<<(BB * HXSZ) / 256, 256, 0, stream>>>(gate, cand, hx1, h1);

  // ---- projection ----
  proj_kernel<<<ROWS / 256, 256, 0, stream>>>(h1, wp, bp, out);
}